// LinearEncoder_21835613733038
// MI455X (gfx1250) — compile-verified
//
#include <hip/hip_runtime.h>

// GCNConv for MI455X (gfx1250): fp32 WMMA GEMM + L2-resident atomic scatter.
// Inputs (setup_inputs order): x[f32 N*256], edge_index[int 2*E], W[f32 256*64], b[f32 64]
// (harness delivers integer inputs as 32-bit ints; N=1e5 < 2^31 so int32 indices are safe)

typedef float v2f __attribute__((ext_vector_type(2)));
typedef float v8f __attribute__((ext_vector_type(8)));

#define CIN 256
#define COUT 64

// ---- degree: exact integer counts (self-loop handled as +1 later) ----------
__global__ void __launch_bounds__(256)
k_zero_deg(unsigned* __restrict__ deg, int N) {
    int i = blockIdx.x * 256 + threadIdx.x;
    if (i < N) deg[i] = 0u;
}

__global__ void __launch_bounds__(256)
k_deg(const int* __restrict__ dstp, unsigned* __restrict__ deg, int E) {
    int e = blockIdx.x * 256 + threadIdx.x;
    if (e < E) atomicAdd(&deg[dstp[e]], 1u);
}

__global__ void __launch_bounds__(256)
k_dinv(const unsigned* __restrict__ deg, float* __restrict__ dinv, int N) {
    int i = blockIdx.x * 256 + threadIdx.x;
    if (i < N) dinv[i] = rsqrtf((float)(deg[i] + 1u)); // +1 = self loop; always > 0
}

// ---- h = X * W via V_WMMA_F32_16X16X4_F32 ----------------------------------
// One wave -> 16x64 output strip (4 n-tiles; A fragment reused across 4 WMMAs).
// W is staged into LDS in *fragment-native* layout:
//   Ws2[p*64 + n] = { W[2p][n], W[2p+1][n] },  p = K/2 (0..127), n = 0..63
// so each B fragment is a single aligned ds_load_b64 into an even VGPR pair
// (WMMA sources must be even-aligned) -> no repack moves in the hot loop.
__global__ void __launch_bounds__(256)
k_gemm(const float* __restrict__ x, const float* __restrict__ W,
       float* __restrict__ h, int N) {
    __shared__ float2 Ws2[(CIN / 2) * COUT]; // 64 KB (WGP has 320 KB LDS)
    for (int idx = threadIdx.x; idx < (CIN / 2) * COUT; idx += 256) {
        int p = idx >> 6, n = idx & 63;
        Ws2[idx] = make_float2(W[p * (2 * COUT) + n], W[p * (2 * COUT) + COUT + n]);
    }
    __syncthreads();

    const int lane = threadIdx.x & 31;
    const int wave = threadIdx.x >> 5;
    const int m0   = (blockIdx.x * 8 + wave) * 16; // 16-row tile base
    const int mr   = lane & 15;                    // M within tile (A) / N within tile (B)
    const int kh   = (lane >> 4) << 1;             // lanes 16-31 hold K+2,K+3

    // Clamp tail rows so WMMA runs with EXEC all-1s; predicate only the store.
    int row  = m0 + mr;
    int rowc = row < N ? row : (N - 1);
    const float* xr = x + (size_t)rowc * CIN;

    v8f acc0 = {}; v8f acc1 = {}; v8f acc2 = {}; v8f acc3 = {};
    for (int k = 0; k < CIN; k += 4) {
        v2f a = *(const v2f*)(xr + k + kh);            // A: K = k+kh, k+kh+1
        const float2* wp = &Ws2[(((k + kh) >> 1) * COUT) + mr];
        v2f b0 = *(const v2f*)(wp + 0);                // cols mr+0  : one ds_load_b64
        v2f b1 = *(const v2f*)(wp + 16);               // cols mr+16
        v2f b2 = *(const v2f*)(wp + 32);               // cols mr+32
        v2f b3 = *(const v2f*)(wp + 48);               // cols mr+48
        acc0 = __builtin_amdgcn_wmma_f32_16x16x4_f32(false, a, false, b0, (short)0, acc0, false, false);
        acc1 = __builtin_amdgcn_wmma_f32_16x16x4_f32(false, a, false, b1, (short)0, acc1, false, false);
        acc2 = __builtin_amdgcn_wmma_f32_16x16x4_f32(false, a, false, b2, (short)0, acc2, false, false);
        acc3 = __builtin_amdgcn_wmma_f32_16x16x4_f32(false, a, false, b3, (short)0, acc3, false, false);
    }

    // C/D layout: VGPR r -> M=r (lanes 0-15) / M=r+8 (lanes 16-31), N=lane&15
    const int nlo = lane & 15;
    const int mhi = (lane >> 4) << 3;
    for (int r = 0; r < 8; ++r) {
        int orow = m0 + r + mhi;
        if (orow < N) {
            float* hp = h + (size_t)orow * COUT + nlo;
            hp[0]  = acc0[r];
            hp[16] = acc1[r];
            hp[32] = acc2[r];
            hp[48] = acc3[r];
        }
    }
}

// ---- out = b + dinv[i]^2 * h[i]  (self-loop term, initializes d_out) -------
__global__ void __launch_bounds__(256)
k_self(const float* __restrict__ h, const float* __restrict__ dinv,
       const float* __restrict__ b, float* __restrict__ out, int N) {
    long long gid = (long long)blockIdx.x * 256 + threadIdx.x;
    int i = (int)(gid >> 6);
    int c = (int)(gid & 63);
    if (i < N) {
        float di = dinv[i];
        out[(size_t)i * COUT + c] = b[c] + di * di * h[(size_t)i * COUT + c];
    }
}

// ---- edge scatter: out[dst] += norm * h[src] -------------------------------
// One wave per edge: edge id is wave-uniform -> scalarize index/degree loads
// into SMEM. Each lane handles channels c and c+32 (coalesced 128B bursts).
// h & out are 25.6 MB each -> resident in the 192 MB L2; f32 atomics resolve
// on-chip at L2 (global_atomic_add_f32 via unsafeAtomicAdd).
__global__ void __launch_bounds__(256)
k_scatter(const int* __restrict__ srcp, const int* __restrict__ dstp,
          const float* __restrict__ dinv, const float* __restrict__ h,
          float* __restrict__ out, int E) {
    long long gid = (long long)blockIdx.x * 256 + threadIdx.x;
    int e = (int)(gid >> 5);       // same for all 32 lanes of a wave
    if (e >= E) return;            // wave-uniform exit
    e = __builtin_amdgcn_readfirstlane(e);
    int lane = threadIdx.x & 31;
    int s = srcp[e];
    int d = dstp[e];
    float norm = dinv[s] * dinv[d];
    const float* hs = h + (size_t)s * COUT;
    float* od = out + (size_t)d * COUT;
    unsafeAtomicAdd(&od[lane],      hs[lane]      * norm);
    unsafeAtomicAdd(&od[lane + 32], hs[lane + 32] * norm);
}

extern "C" void kernel_launch(void* const* d_in, const int* in_sizes, int n_in,
                              void* d_out, int out_size, void* d_ws, size_t ws_size,
                              hipStream_t stream) {
    const float* x  = (const float*)d_in[0];
    const int*   ei = (const int*)d_in[1];
    const float* W  = (const float*)d_in[2];
    const float* b  = (const float*)d_in[3];
    float* out = (float*)d_out;

    const int N = in_sizes[0] / CIN;  // 100000
    const int E = in_sizes[1] / 2;    // 3200000
    const int* srcp = ei;             // edge_index[0]
    const int* dstp = ei + E;         // edge_index[1]

    // workspace: h [N*64 f32] | deg [N u32] | dinv [N f32]  (~26.4 MB)
    char* ws = (char*)d_ws;
    float*    h    = (float*)ws;
    unsigned* deg  = (unsigned*)(ws + (size_t)N * COUT * sizeof(float));
    float*    dinv = (float*)(ws + (size_t)N * COUT * sizeof(float) + (size_t)N * sizeof(unsigned));

    int nb;
    nb = (N + 255) / 256;
    k_zero_deg<<<nb, 256, 0, stream>>>(deg, N);
    nb = (E + 255) / 256;
    k_deg<<<nb, 256, 0, stream>>>(dstp, deg, E);
    nb = (N + 255) / 256;
    k_dinv<<<nb, 256, 0, stream>>>(deg, dinv, N);

    nb = (N + 127) / 128; // 8 waves/block, 16 rows/wave
    k_gemm<<<nb, 256, 0, stream>>>(x, W, h, N);

    long long t = (long long)N * COUT;
    nb = (int)((t + 255) / 256);
    k_self<<<nb, 256, 0, stream>>>(h, dinv, b, out, N);

    t = (long long)E * 32;
    nb = (int)((t + 255) / 256);
    k_scatter<<<nb, 256, 0, stream>>>(srcp, dstp, dinv, h, out, E);
}